// LSSViewTransformer_29626684408006
// MI455X (gfx1250) — compile-verified
//
#include <hip/hip_runtime.h>
#include <math.h>

// ---- problem constants (from reference) ----
#define FH      32
#define FW      88
#define NPIX    (FH * FW)      // 2816
#define CIN     256
#define DBINS   41
#define CCH     64
#define OCH     105            // D + C
#define OPAD    112            // padded to 7*16
#define NCAM    6
#define NVOX    16384          // 128*128
#define NTM     7              // M tiles of 16
#define COLT    (NPIX / 16)    // 176 column tiles
#define WPB     4              // waves per block

typedef float v2f __attribute__((ext_vector_type(2)));
typedef float v8f __attribute__((ext_vector_type(8)));

__global__ void lss_zero_kernel(float* __restrict__ out, int n) {
    int i = blockIdx.x * blockDim.x + threadIdx.x;
    int stride = gridDim.x * blockDim.x;
    for (; i < n; i += stride) out[i] = 0.0f;
}

__device__ __forceinline__ void inv3(const float* __restrict__ m, float* o) {
    float a = m[0], b = m[1], c = m[2];
    float d = m[3], e = m[4], f = m[5];
    float g = m[6], h = m[7], i = m[8];
    float A =  (e * i - f * h);
    float B = -(d * i - f * g);
    float C =  (d * h - e * g);
    float det = a * A + b * B + c * C;
    float r = 1.0f / det;
    o[0] = A * r;               o[1] = -(b * i - c * h) * r;  o[2] =  (b * f - c * e) * r;
    o[3] = B * r;               o[4] =  (a * i - c * g) * r;  o[5] = -(a * f - c * d) * r;
    o[6] = C * r;               o[7] = -(a * h - b * g) * r;  o[8] =  (a * e - b * d) * r;
}

__global__ __launch_bounds__(WPB * 32) void lss_view_transform_kernel(
    const float* __restrict__ x,          // (6, 256, 32, 88)
    const float* __restrict__ rots,       // (1, 6, 3, 3)
    const float* __restrict__ trans,      // (1, 6, 3)
    const float* __restrict__ intrins,    // (1, 6, 3, 3)
    const float* __restrict__ post_rots,  // (1, 6, 3, 3)
    const float* __restrict__ post_trans, // (1, 6, 3)
    const float* __restrict__ w_depth,    // (105, 256)
    const float* __restrict__ b_depth,    // (105,)
    float* __restrict__ out)              // (64, 128, 128) flat: c*16384 + rank
{
    __shared__ float lds[WPB * 16 * OPAD];

    const int lane  = threadIdx.x & 31;
    const int wib   = threadIdx.x >> 5;
    const int wid   = blockIdx.x * WPB + wib;
    const int n     = wid / COLT;        // camera
    const int ct    = wid % COLT;        // column tile
    const int colBase = ct * 16;
    const int l15   = lane & 15;
    const int khalf = lane >> 4;
    float* sl = lds + wib * (16 * OPAD);

    // ---------------- GEMM: feats tile (112 x 16) via fp32 WMMA ----------------
    v8f acc[NTM];
    #pragma unroll
    for (int mt = 0; mt < NTM; ++mt)
        #pragma unroll
        for (int r = 0; r < 8; ++r) acc[mt][r] = 0.0f;

    const float* xb = x + n * (CIN * NPIX) + colBase + l15;

    #pragma unroll 2
    for (int k0 = 0; k0 < CIN; k0 += 4) {
        const int k = k0 + 2 * khalf;           // B rows k, k+1 for this lane half
        v2f bfrag;
        bfrag.x = xb[k * NPIX];
        bfrag.y = xb[(k + 1) * NPIX];
        #pragma unroll
        for (int mt = 0; mt < NTM; ++mt) {
            const int m  = mt * 16 + l15;
            const int mc = (m < OCH) ? m : 0;   // clamp: keep loads in-bounds
            v2f afrag;
            afrag.x = w_depth[mc * CIN + k];
            afrag.y = w_depth[mc * CIN + k + 1];
            if (m >= OCH) { afrag.x = 0.0f; afrag.y = 0.0f; }
            acc[mt] = __builtin_amdgcn_wmma_f32_16x16x4_f32(
                false, afrag, false, bfrag, (short)0, acc[mt], false, false);
        }
    }

    // Store tile to LDS: lane owns 8 consecutive m rows per mt (M = mt*16 + 8*khalf + r)
    #pragma unroll
    for (int mt = 0; mt < NTM; ++mt) {
        float* p = sl + l15 * OPAD + mt * 16 + 8 * khalf;
        #pragma unroll
        for (int r = 0; r < 8; ++r) p[r] = acc[mt][r];
    }
    __syncthreads();

    // ---------------- per-camera geometry matrices (uniform per wave) ----------------
    float Kinv[9], Pinv[9], comb[9];
    inv3(intrins + n * 9, Kinv);
    inv3(post_rots + n * 9, Pinv);
    const float* R = rots + n * 9;
    #pragma unroll
    for (int i = 0; i < 3; ++i)
        #pragma unroll
        for (int j = 0; j < 3; ++j)
            comb[i * 3 + j] = R[i * 3 + 0] * Kinv[0 + j]
                            + R[i * 3 + 1] * Kinv[3 + j]
                            + R[i * 3 + 2] * Kinv[6 + j];
    const float tx  = trans[n * 3 + 0], ty  = trans[n * 3 + 1], tz  = trans[n * 3 + 2];
    const float ptx = post_trans[n * 3 + 0], pty = post_trans[n * 3 + 1], ptz = post_trans[n * 3 + 2];

    // ---------------- per-lane biases (all indices in-bounds for 105 entries) ----------------
    const float bd0 = b_depth[lane];        // depth bin d = lane
    const float bd1 = b_depth[32 + lane];   // depth bin d = lane+32 (used only lane<9)
    const float bc0 = b_depth[41 + lane];   // ctx channel c = lane
    const float bc1 = b_depth[73 + lane];   // ctx channel c = lane+32

    // ---------------- softmax + geometry + scatter, one pixel (column) at a time ----------------
    for (int col = 0; col < 16; ++col) {
        const float* fs = sl + col * OPAD;

        // depth logits for d = lane, lane+32
        float L0 = fs[lane] + bd0;
        float L1 = (lane < 9) ? (fs[32 + lane] + bd1) : -INFINITY;
        float mx = fmaxf(L0, L1);
        #pragma unroll
        for (int off = 16; off >= 1; off >>= 1)
            mx = fmaxf(mx, __shfl_xor(mx, off, 32));
        float e0 = expf(L0 - mx);
        float e1 = (lane < 9) ? expf(L1 - mx) : 0.0f;
        float s  = e0 + e1;
        #pragma unroll
        for (int off = 16; off >= 1; off >>= 1)
            s += __shfl_xor(s, off, 32);
        const float sinv = 1.0f / s;

        // context channels c = lane, lane+32
        const float c0 = fs[41 + lane] + bc0;
        const float c1 = fs[73 + lane] + bc1;

        // frustum pixel -> camera ray (affine in depth ds)
        const int pix = colBase + col;
        const int hh = pix / FW, ww = pix % FW;
        const float xs = (float)ww * (1407.0f / 87.0f);
        const float ys = (float)hh * (511.0f / 31.0f);
        const float u = xs - ptx, v = ys - pty;
        const float qbx = Pinv[0] * u + Pinv[1] * v - Pinv[2] * ptz;
        const float qby = Pinv[3] * u + Pinv[4] * v - Pinv[5] * ptz;
        const float qbz = Pinv[6] * u + Pinv[7] * v - Pinv[8] * ptz;

        // voxel ranks for d = lane and d = lane+32 (computed in parallel, broadcast later)
        int rnk[2];
        #pragma unroll
        for (int half = 0; half < 2; ++half) {
            const float ds = 4.0f + (float)(lane + 32 * half);
            const float qx = qbx + Pinv[2] * ds;
            const float qy = qby + Pinv[5] * ds;
            const float qz = qbz + Pinv[8] * ds;
            const float px = qx * qz, py = qy * qz, pz = qz;
            const float wx = comb[0] * px + comb[1] * py + comb[2] * pz + tx;
            const float wy = comb[3] * px + comb[4] * py + comb[5] * pz + ty;
            const float wz = comb[6] * px + comb[7] * py + comb[8] * pz + tz;
            const int gx = (int)((wx + 51.2f) / 0.8f);   // trunc-toward-zero, as astype(int32)
            const int gy = (int)((wy + 51.2f) / 0.8f);
            const int gz = (int)((wz + 10.0f) / 20.0f);
            rnk[half] = gx + gy * 128 + gz * NVOX;
        }

        // scatter: ranks along a ray are wave-uniform; run-length-merge consecutive
        // depth bins that land in the same voxel to cut global atomic traffic
        // (atomics into the L2-resident BEV grid are the workload's bottleneck).
        int   prevRk = -1;      // -1 is never a kept rank, safe sentinel
        float wacc   = 0.0f;
        for (int d = 0; d < DBINS; ++d) {
            const float ew = (d < 32) ? e0 : e1;
            const int   rw = (d < 32) ? rnk[0] : rnk[1];
            const float wgt = __shfl(ew, d & 31, 32) * sinv;
            const int   rk  = __shfl(rw, d & 31, 32);
            if (rk == prevRk) {                 // wave-uniform branch
                wacc += wgt;
            } else {
                if (prevRk >= 0 && prevRk < NVOX) {
                    atomicAdd(out + lane * NVOX + prevRk,        c0 * wacc);
                    atomicAdd(out + (lane + 32) * NVOX + prevRk, c1 * wacc);
                }
                prevRk = rk;
                wacc   = wgt;
            }
        }
        if (prevRk >= 0 && prevRk < NVOX) {
            atomicAdd(out + lane * NVOX + prevRk,        c0 * wacc);
            atomicAdd(out + (lane + 32) * NVOX + prevRk, c1 * wacc);
        }
    }
}

extern "C" void kernel_launch(void* const* d_in, const int* in_sizes, int n_in,
                              void* d_out, int out_size, void* d_ws, size_t ws_size,
                              hipStream_t stream) {
    const float* x          = (const float*)d_in[0];
    const float* rots       = (const float*)d_in[1];
    const float* trans      = (const float*)d_in[2];
    const float* intrins    = (const float*)d_in[3];
    const float* post_rots  = (const float*)d_in[4];
    const float* post_trans = (const float*)d_in[5];
    const float* w_depth    = (const float*)d_in[6];
    const float* b_depth    = (const float*)d_in[7];
    float* out = (float*)d_out;

    // harness poisons d_out; we accumulate with atomics, so zero it first
    lss_zero_kernel<<<512, 256, 0, stream>>>(out, out_size);

    const int nwaves  = NCAM * COLT;        // 1056 tiles, one wave each
    const int nblocks = nwaves / WPB;       // 264 (exact)
    lss_view_transform_kernel<<<nblocks, WPB * 32, 0, stream>>>(
        x, rots, trans, intrins, post_rots, post_trans, w_depth, b_depth, out);
}